// meanAveragePrecisionLoss_27178553049478
// MI455X (gfx1250) — compile-verified
//
#include <hip/hip_runtime.h>

#define B_DIM 512
#define L_DIM 100
#define DATA_LEN 100000
#define MARGIN_F 1.0f
#define GAMMA_F 0.9f

typedef float v2f __attribute__((ext_vector_type(2)));
typedef float v8f __attribute__((ext_vector_type(8)));

// One block per label l. 512 threads = 16 waves; wave w owns i-rows
// [32w, 32w+32). Row sums over j computed with V_WMMA_F32_16X16X4_F32:
//   A = 16x4 tile of sur values, B column0 = 1 (-> S_all), column1 = pos_j
//   (-> S_pos), other columns 0. One WMMA per tile yields both reductions.
__global__ __launch_bounds__(512) void map_loss_main(
    const float* __restrict__ y_pred,   // (B, L)
    const float* __restrict__ y_true,   // (B, L)
    const int*   __restrict__ index,    // (B,)
    const float* __restrict__ u_all,    // (L, DATA_LEN, 1)
    const float* __restrict__ u_pos,    // (L, DATA_LEN, 1)
    float*       __restrict__ per_label) // (L,)
{
    __shared__ float s_fp[B_DIM];
    __shared__ float s_pos[B_DIM];
    __shared__ float s_sall[B_DIM];
    __shared__ float s_spos[B_DIM];
    __shared__ float s_r1[B_DIM];
    __shared__ float s_r2[B_DIM];

    const int l = blockIdx.x;
    const int t = threadIdx.x;

    // Stage fp[l, :] = y_pred[:, l] and pos[l, :] into LDS.
    s_fp[t]  = y_pred[(size_t)t * L_DIM + l];
    s_pos[t] = (y_true[(size_t)t * L_DIM + l] == 1.0f) ? 1.0f : 0.0f;
    __syncthreads();

    const int lane  = t & 31;
    const int wave  = t >> 5;
    const int m     = lane & 15;          // A row within 16-row group
    const int kbase = (lane >> 4) << 1;   // lanes 0-15: K=0,1; lanes 16-31: K=2,3
    const int n     = m;                  // B/D column id

    const float fpi0 = s_fp[wave * 32 + m];        // group 2w
    const float fpi1 = s_fp[wave * 32 + 16 + m];   // group 2w+1

    v8f acc0 = {};
    v8f acc1 = {};

    for (int jb = 0; jb < B_DIM; jb += 4) {
        const float fj0 = s_fp[jb + kbase];
        const float fj1 = s_fp[jb + kbase + 1];
        const float p0  = s_pos[jb + kbase];
        const float p1  = s_pos[jb + kbase + 1];

        const float d00 = fmaxf(MARGIN_F - (fpi0 - fj0), 0.0f);
        const float d01 = fmaxf(MARGIN_F - (fpi0 - fj1), 0.0f);
        const float d10 = fmaxf(MARGIN_F - (fpi1 - fj0), 0.0f);
        const float d11 = fmaxf(MARGIN_F - (fpi1 - fj1), 0.0f);

        v2f a0; a0.x = d00 * d00; a0.y = d01 * d01;
        v2f a1; a1.x = d10 * d10; a1.y = d11 * d11;
        // B[k,0] = 1, B[k,1] = pos[jb+k], else 0  (per-lane build, no branches)
        v2f b;
        b.x = (n == 0) ? 1.0f : ((n == 1) ? p0 : 0.0f);
        b.y = (n == 0) ? 1.0f : ((n == 1) ? p1 : 0.0f);

        acc0 = __builtin_amdgcn_wmma_f32_16x16x4_f32(false, a0, false, b,
                                                     (short)0, acc0, false, false);
        acc1 = __builtin_amdgcn_wmma_f32_16x16x4_f32(false, a1, false, b,
                                                     (short)0, acc1, false, false);
    }

    // Extract D columns 0 (S_all) and 1 (S_pos). VGPR v holds row v (+8 for
    // lanes 16-31); column = lane & 15.
    if (n < 2) {
        const int half = (lane >> 4) * 8;
        #pragma unroll
        for (int v = 0; v < 8; ++v) {
            const int r0 = wave * 32 + half + v;
            const int r1 = wave * 32 + 16 + half + v;
            if (n == 0) { s_sall[r0] = acc0[v]; s_sall[r1] = acc1[v]; }
            else        { s_spos[r0] = acc0[v]; s_spos[r1] = acc1[v]; }
        }
    }
    __syncthreads();

    // Per-i epilogue (EMA update + contribution).
    {
        const float S_all = s_sall[t];
        const float S_pos = s_spos[t];
        const float posi  = s_pos[t];
        const int   idx   = index[t];
        const float ua = u_all[(size_t)l * DATA_LEN + idx];
        const float up = u_pos[(size_t)l * DATA_LEN + idx];
        const float rm  = S_all * (1.0f / (float)B_DIM);   // row_mean
        const float prm = S_pos * (1.0f / (float)B_DIM);   // pos_row_mean
        const float ua_new = (posi > 0.0f) ? ((1.0f - GAMMA_F) * ua + GAMMA_F * rm)  : ua;
        const float up_new = (posi > 0.0f) ? ((1.0f - GAMMA_F) * up + GAMMA_F * prm) : up;
        const float denom  = (posi > 0.0f) ? ua_new : 1.0f;
        // sum_j p_i*sur = (up_new*S_all - ua_new*S_pos)/denom^2
        const float contrib = posi * (up_new * S_all - ua_new * S_pos) / (denom * denom);
        s_r1[t] = contrib;
        s_r2[t] = posi;   // for n_pos
    }
    __syncthreads();

    for (int s = B_DIM / 2; s > 0; s >>= 1) {
        if (t < s) { s_r1[t] += s_r1[t + s]; s_r2[t] += s_r2[t + s]; }
        __syncthreads();
    }
    if (t == 0) {
        per_label[l] = s_r1[0] / (s_r2[0] * (float)B_DIM);
    }
}

__global__ __launch_bounds__(128) void map_loss_reduce(
    const float* __restrict__ per_label, float* __restrict__ out)
{
    __shared__ float s[128];
    const int t = threadIdx.x;
    s[t] = (t < L_DIM) ? per_label[t] : 0.0f;
    __syncthreads();
    for (int step = 64; step > 0; step >>= 1) {
        if (t < step) s[t] += s[t + step];
        __syncthreads();
    }
    if (t == 0) out[0] = s[0] * (1.0f / (float)L_DIM);
}

extern "C" void kernel_launch(void* const* d_in, const int* in_sizes, int n_in,
                              void* d_out, int out_size, void* d_ws, size_t ws_size,
                              hipStream_t stream) {
    (void)in_sizes; (void)n_in; (void)out_size; (void)ws_size;
    const float* y_pred = (const float*)d_in[0];
    const float* y_true = (const float*)d_in[1];
    const int*   index  = (const int*)d_in[2];
    const float* u_all  = (const float*)d_in[3];
    const float* u_pos  = (const float*)d_in[4];
    float* per_label = (float*)d_ws;    // 100 floats of scratch
    float* out       = (float*)d_out;

    map_loss_main<<<L_DIM, 512, 0, stream>>>(y_pred, y_true, index, u_all, u_pos, per_label);
    map_loss_reduce<<<1, 128, 0, stream>>>(per_label, out);
}